// Translator_31499290149149
// MI455X (gfx1250) — compile-verified
//
#include <hip/hip_runtime.h>
#include <math.h>

#define N_ENT   40000
#define NUM_POS 24
#define N_RELS  48
#define BODYLEN 3
#define N_EDGES 300000
#define BATCH   64
#define THRV    1e-20f

typedef float v2f __attribute__((ext_vector_type(2)));
typedef float v8f __attribute__((ext_vector_type(8)));

__device__ __forceinline__ void atomAddF(float* p, float v) {
    unsafeAtomicAdd(p, v);   // hardware global_atomic_add_f32 on gfx1250
}

// ---------------------------------------------------------------------------
// K1: vals[e] = killed(e) ? 0 : edge_val[e]
// killed if ANY batch row matches (head, tail, rel) == (q_head, t, q_rel).
// ---------------------------------------------------------------------------
__global__ void k_vals(const int* __restrict__ eh, const int* __restrict__ et,
                       const int* __restrict__ er, const float* __restrict__ ev,
                       const int* __restrict__ qh, const int* __restrict__ qr,
                       const int* __restrict__ qt, float* __restrict__ vals) {
    __shared__ int sh[BATCH], sr[BATCH], st[BATCH];
    int tid = threadIdx.x;
    if (tid < BATCH) { sh[tid] = qh[tid]; sr[tid] = qr[tid]; st[tid] = qt[tid]; }
    __syncthreads();
    int e = blockIdx.x * blockDim.x + tid;
    if (e >= N_EDGES) return;
    int h = eh[e], t = et[e], r = er[e];
    bool killed = false;
#pragma unroll 8
    for (int b = 0; b < BATCH; ++b)
        killed |= (h == sh[b]) & (t == st[b]) & (r == sr[b]);
    vals[e] = killed ? 0.0f : ev[e];
}

// ---------------------------------------------------------------------------
// K2a: zero a float buffer
// ---------------------------------------------------------------------------
__global__ void k_zero(float* __restrict__ p, int n) {
    int i = blockIdx.x * blockDim.x + threadIdx.x;
    if (i < n) p[i] = 0.0f;
}

// K2b: one-hot seed of mem (layout [ent][batch]) + zero the sums accumulator
__global__ void k_onehot(const int* __restrict__ qh, float* __restrict__ W0,
                         float* __restrict__ sums) {
    int b = threadIdx.x;                       // 64 threads
    sums[b] = 0.0f;
    W0[(size_t)qh[b] * BATCH + b] = 1.0f;
}

// ---------------------------------------------------------------------------
// K3a: Bm[n][b] = A[n][b] * attention[b, step, 47]   (self-loop term = init)
// ---------------------------------------------------------------------------
__global__ void k_selfinit(const float* __restrict__ A, float* __restrict__ Bm,
                           const float* __restrict__ attention, int step) {
    int i = blockIdx.x * blockDim.x + threadIdx.x;
    if (i >= N_ENT * BATCH) return;
    int b = i & (BATCH - 1);
    float self = attention[b * (BODYLEN * N_RELS) + step * N_RELS + (N_RELS - 1)];
    Bm[i] = A[i] * self;
}

// ---------------------------------------------------------------------------
// K3b: edge scatter. 64 threads (2 waves) per edge: lane b handles batch b.
// Contiguous 256B gathers/scatters in the [ent][batch] layout; attention
// table (64x48 = 12KB) staged in LDS; zero contributions skip the atomic.
// ---------------------------------------------------------------------------
__global__ void k_edges(const int* __restrict__ eh, const int* __restrict__ et,
                        const int* __restrict__ er, const float* __restrict__ vals,
                        const float* __restrict__ attention, int step,
                        const float* __restrict__ A, float* __restrict__ Bm) {
    __shared__ float sat[BATCH * N_RELS];
    for (int i = threadIdx.x; i < BATCH * N_RELS; i += blockDim.x) {
        int b = i / N_RELS, r = i % N_RELS;
        sat[i] = attention[b * (BODYLEN * N_RELS) + step * N_RELS + r];
    }
    __syncthreads();
    int g = blockIdx.x * blockDim.x + threadIdx.x;
    int e = g >> 6;
    int b = g & 63;
    if (e >= N_EDGES) return;
    float v = vals[e];
    if (v == 0.0f) return;
    int h = eh[e], t = et[e], r = er[e];
    float mh = A[h * BATCH + b];
    float mt = A[t * BATCH + b];
    float c1 = mh * v * sat[b * N_RELS + r];            // fwd: h -> t
    float c2 = mt * v * sat[b * N_RELS + r + NUM_POS];  // rev: t -> h
    if (c1 != 0.0f) atomAddF(&Bm[t * BATCH + b], c1);
    if (c2 != 0.0f) atomAddF(&Bm[h * BATCH + b], c2);
}

// ---------------------------------------------------------------------------
// K4: per-batch row sums via V_WMMA_F32_16X16X4_F32.
// A = 16x4 ones, B = 4 entities x 16 batch cols  =>  D[m,n] += sum_k B[k,n].
// 4 waves per block cover the 64 batch columns; each wave accumulates its
// entity chunk in the f32 C operand, then lanes 0-15 fold c[0] into sums.
// ---------------------------------------------------------------------------
__global__ void k_sums_wmma(const float* __restrict__ W, float* __restrict__ sums) {
    int lane = threadIdx.x & 31;
    int wv   = threadIdx.x >> 5;                 // wave 0..3 -> batch cols
    int col  = (lane & 15) + (wv << 4);
    int kb   = (lane >> 4) << 1;                 // lanes 0-15: K{0,1}; 16-31: K{2,3}
    v2f a; a.x = 1.0f; a.y = 1.0f;               // ones A-matrix
    v8f c = {};
    for (int e0 = blockIdx.x * 4; e0 < N_ENT; e0 += gridDim.x * 4) {
        v2f bv;
        bv.x = W[(size_t)(e0 + kb)     * BATCH + col];
        bv.y = W[(size_t)(e0 + kb + 1) * BATCH + col];
        c = __builtin_amdgcn_wmma_f32_16x16x4_f32(false, a, false, bv,
                                                  (short)0, c, false, false);
    }
    if (lane < 16) atomAddF(&sums[col], c[0]);
}

// ---------------------------------------------------------------------------
// K5: loss = -mean_b log(max(THR, mem[b, t[b]] / max(THR, sums[b])))
// ---------------------------------------------------------------------------
__global__ void k_loss(const float* __restrict__ W, const float* __restrict__ sums,
                       const int* __restrict__ qt, float* __restrict__ out) {
    __shared__ float red[BATCH];
    int b = threadIdx.x;
    float s = fmaxf(THRV, sums[b]);
    float p = W[(size_t)qt[b] * BATCH + b] / s;
    red[b] = -logf(fmaxf(THRV, p));
    __syncthreads();
    for (int off = 32; off > 0; off >>= 1) {
        if (b < off) red[b] += red[b + off];
        __syncthreads();
    }
    if (b == 0) out[0] = red[0] / (float)BATCH;
}

// ---------------------------------------------------------------------------
// K6: normalize + transpose [ent][batch] -> d_out[1 + b*N_ENT + n]
// ---------------------------------------------------------------------------
__global__ void k_norm(const float* __restrict__ W, const float* __restrict__ sums,
                       float* __restrict__ out) {
    int g = blockIdx.x * blockDim.x + threadIdx.x;
    if (g >= N_ENT * BATCH) return;
    int b = g / N_ENT;
    int n = g - b * N_ENT;
    out[1 + g] = W[(size_t)n * BATCH + b] / fmaxf(THRV, sums[b]);
}

// ---------------------------------------------------------------------------
extern "C" void kernel_launch(void* const* d_in, const int* in_sizes, int n_in,
                              void* d_out, int out_size, void* d_ws, size_t ws_size,
                              hipStream_t stream) {
    const int*   q_head = (const int*)  d_in[0];
    const int*   q_rel  = (const int*)  d_in[1];
    const int*   q_t    = (const int*)  d_in[2];
    const float* attn   = (const float*)d_in[3];
    const int*   e_head = (const int*)  d_in[4];
    const int*   e_tail = (const int*)  d_in[5];
    const int*   e_rel  = (const int*)  d_in[6];
    const float* e_val  = (const float*)d_in[7];
    float* out = (float*)d_out;

    // workspace carve (~21.7 MB)
    float* W0   = (float*)d_ws;
    float* W1   = W0 + (size_t)N_ENT * BATCH;
    float* vals = W1 + (size_t)N_ENT * BATCH;
    float* sums = vals + N_EDGES;

    const int MEMN = N_ENT * BATCH;                    // 2,560,000

    // 1. kill mask -> vals
    k_vals<<<(N_EDGES + 255) / 256, 256, 0, stream>>>(
        e_head, e_tail, e_rel, e_val, q_head, q_rel, q_t, vals);

    // 2. mem := one_hot(q_head) in [ent][batch] layout; sums := 0
    k_zero<<<(MEMN + 255) / 256, 256, 0, stream>>>(W0, MEMN);
    k_onehot<<<1, BATCH, 0, stream>>>(q_head, W0, sums);

    // 3. three propagation steps, ping-pong W0/W1
    float* A  = W0;
    float* Bm = W1;
    for (int step = 0; step < BODYLEN; ++step) {
        k_selfinit<<<(MEMN + 255) / 256, 256, 0, stream>>>(A, Bm, attn, step);
        k_edges<<<(N_EDGES * BATCH) / 256, 256, 0, stream>>>(
            e_head, e_tail, e_rel, vals, attn, step, A, Bm);
        float* tmp = A; A = Bm; Bm = tmp;
    }
    // final mem now in A

    // 4. row sums via WMMA
    k_sums_wmma<<<256, 128, 0, stream>>>(A, sums);

    // 5. loss and normalized output
    k_loss<<<1, BATCH, 0, stream>>>(A, sums, q_t, out);
    k_norm<<<(MEMN + 255) / 256, 256, 0, stream>>>(A, sums, out);
}